// ClassLayer_25658134626613
// MI455X (gfx1250) — compile-verified
//
#include <hip/hip_runtime.h>

#define NN 262144
#define DD 128
#define KK 512
#define GG 4096
#define TT 10

// LDS row stride (floats) for the staged codebook: 132 % 64 = 4 banks apart per
// consecutive code row -> the 16 lanes of a B-fragment 8-byte load hit 32
// distinct banks (conflict-free); keeps 8-byte alignment.
#define LDSTRIDE 132

typedef __attribute__((ext_vector_type(2))) float v2f;
typedef __attribute__((ext_vector_type(8))) float v8f;

// ---------------------------------------------------------------- c2 = ||codebook_row||^2
__global__ __launch_bounds__(128) void c2_kernel(const float* __restrict__ cb,
                                                 float* __restrict__ c2) {
    int k = blockIdx.x * blockDim.x + threadIdx.x;
    if (k < KK) {
        const float* r = cb + (size_t)k * DD;
        float s = 0.f;
#pragma unroll 8
        for (int d = 0; d < DD; ++d) s += r[d] * r[d];
        c2[k] = s;
    }
}

// ---------------------------------------------------------------- WMMA argmin over codes
// grid = NN/128 blocks, 256 threads (8 waves); each wave handles 16 rows of x.
// Full causal codebook staged in LDS (512 * 132 * 4B = 264KB, < 320KB WGP LDS).
__global__ __launch_bounds__(256) void argmin_kernel(const float* __restrict__ x,
                                                     const float* __restrict__ cb,
                                                     const float* __restrict__ c2,
                                                     int* __restrict__ indices) {
    extern __shared__ float lds[];
    const int tid = threadIdx.x;

    // cooperative stage: 512 rows x 64 float2
    for (int i = tid; i < KK * (DD / 2); i += 256) {
        int code = i >> 6;
        int d2i  = i & 63;
        *(v2f*)&lds[code * LDSTRIDE + d2i * 2] =
            *(const v2f*)&cb[(size_t)code * DD + d2i * 2];
    }
    __syncthreads();

    const int wave = tid >> 5;
    const int lane = tid & 31;
    const int m    = lane & 15;   // A-row / B-col within 16x16 tile
    const int half = lane >> 4;   // K-pair selector for x4 fragments
    const size_t row0 = (size_t)blockIdx.x * 128 + (size_t)wave * 16;
    const float* xr = x + (row0 + m) * DD + 2 * half;   // A-frag base for this lane

    float bestv[8];
    int   besti[8];
#pragma unroll
    for (int v = 0; v < 8; ++v) { bestv[v] = 3.4e38f; besti[v] = 0; }

    for (int g = 0; g < 4; ++g) {          // 4 groups x 128 codes
        v8f acc[8] = {};                   // 8 N-tiles of 16 codes each
        const float* ldsg = lds + (g * 128 + m) * LDSTRIDE + 2 * half;
        for (int kb = 0; kb < DD; kb += 4) {
            v2f a = *(const v2f*)(xr + kb);            // A[m][kb+2h .. +1]
#pragma unroll
            for (int t = 0; t < 8; ++t) {
                v2f b = *(const v2f*)(ldsg + t * 16 * LDSTRIDE + kb); // cb[n][kb+2h..]
                acc[t] = __builtin_amdgcn_wmma_f32_16x16x4_f32(
                    false, a, false, b, (short)0, acc[t], false, false);
            }
        }
        // fold c2 - 2*dot, update per-lane argmin (each lane owns column n)
#pragma unroll
        for (int t = 0; t < 8; ++t) {
            int n = g * 128 + t * 16 + m;
            float cn = c2[n];
#pragma unroll
            for (int v = 0; v < 8; ++v) {
                float d = cn - 2.f * acc[t][v];
                if (d < bestv[v] || (d == bestv[v] && n < besti[v])) {
                    bestv[v] = d; besti[v] = n;
                }
            }
        }
    }

    // reduce across the 16 lanes of each half (xor masks 1,2,4,8 stay in-half)
#pragma unroll
    for (int mask = 1; mask <= 8; mask <<= 1) {
#pragma unroll
        for (int v = 0; v < 8; ++v) {
            float ov = __shfl_xor(bestv[v], mask, 32);
            int   oi = __shfl_xor(besti[v], mask, 32);
            if (ov < bestv[v] || (ov == bestv[v] && oi < besti[v])) {
                bestv[v] = ov; besti[v] = oi;
            }
        }
    }
    // lanes 0..15 hold rows 0..7, lanes 16..31 hold rows 8..15 (C layout)
    if (m == 0) {
#pragma unroll
        for (int v = 0; v < 8; ++v) indices[row0 + 8 * half + v] = besti[v];
    }
}

// ---------------------------------------------------------------- per-graph pooling + gather
__global__ __launch_bounds__(128) void pool_kernel(const float* __restrict__ x,
                                                   const long long* __restrict__ batch,
                                                   const float* __restrict__ codebook,
                                                   const float* __restrict__ causal_cb,
                                                   const float* __restrict__ counter_cb,
                                                   const int* __restrict__ indices,
                                                   float* __restrict__ z_nodes,
                                                   float* __restrict__ pooled_causal,
                                                   float* __restrict__ pooled_x,
                                                   float* __restrict__ pooled_counter) {
    const int g = blockIdx.x;
    const int t = threadIdx.x;

    // lower_bound(batch, g) and lower_bound(batch, g+1) on sorted int64 batch
    int lo = 0, hi = NN;
    while (lo < hi) { int mid = (lo + hi) >> 1; if (batch[mid] < (long long)g) lo = mid + 1; else hi = mid; }
    const int start = lo;
    hi = NN;
    while (lo < hi) { int mid = (lo + hi) >> 1; if (batch[mid] < (long long)g + 1) lo = mid + 1; else hi = mid; }
    const int end = lo;

    float sx = 0.f, sca = 0.f, sco = 0.f;
    for (int i = start; i < end; ++i) {
        int idx = indices[i];                       // uniform across block -> scalar load
        sx  += x[(size_t)i * DD + t];
        sca += causal_cb[(size_t)idx * DD + t];
        sco += counter_cb[(size_t)idx * DD + t];
        z_nodes[(size_t)i * DD + t] = codebook[(size_t)idx * DD + t];
    }
    const float denom = (end > start) ? (float)(end - start) : 1.f;
    pooled_x[(size_t)g * DD + t]       = sx / denom;
    pooled_causal[(size_t)g * DD + t]  = (sx + sca) / denom;   // x + selected_causal pooled
    pooled_counter[(size_t)g * DD + t] = sco / denom;
}

// ---------------------------------------------------------------- tiny FC heads (T=10)
__global__ __launch_bounds__(128) void fc_kernel(const float* __restrict__ pooled_causal,
                                                 const float* __restrict__ pooled_counter,
                                                 const float* __restrict__ pooled_x,
                                                 const float* __restrict__ w,
                                                 const float* __restrict__ b,
                                                 float* __restrict__ causal_pre,
                                                 float* __restrict__ counter_pre,
                                                 float* __restrict__ y_pre) {
    int id = blockIdx.x * blockDim.x + threadIdx.x;
    if (id >= GG * TT) return;
    int g = id / TT, t = id - g * TT;
    const float* wr = w + t * DD;
    const float* pc = pooled_causal  + (size_t)g * DD;
    const float* pn = pooled_counter + (size_t)g * DD;
    const float* px = pooled_x       + (size_t)g * DD;
    float s1 = 0.f, s2 = 0.f, s3 = 0.f;
#pragma unroll 4
    for (int d = 0; d < DD; ++d) {
        float wv = wr[d];
        s1 += pc[d] * wv;
        s2 += pn[d] * wv;
        s3 += px[d] * wv;
    }
    float bb = b[t];
    causal_pre[id]  = s1 + bb;
    counter_pre[id] = s2 + bb;
    y_pre[id]       = s3 + bb;
}

extern "C" void kernel_launch(void* const* d_in, const int* in_sizes, int n_in,
                              void* d_out, int out_size, void* d_ws, size_t ws_size,
                              hipStream_t stream) {
    const float*     x          = (const float*)d_in[0];
    const long long* batch      = (const long long*)d_in[1];
    const float*     codebook   = (const float*)d_in[2];
    const float*     causal_cb  = (const float*)d_in[3];
    const float*     counter_cb = (const float*)d_in[4];
    const float*     fc_w       = (const float*)d_in[5];
    const float*     fc_b       = (const float*)d_in[6];

    // outputs concatenated flat in tuple order
    float* out           = (float*)d_out;
    float* causal_pre    = out;                                   // G*T
    float* counter_pre   = causal_pre + (size_t)GG * TT;          // G*T
    float* y_pre         = counter_pre + (size_t)GG * TT;         // G*T
    float* z_nodes       = y_pre + (size_t)GG * TT;               // N*D
    float* pooled_causal = z_nodes + (size_t)NN * DD;             // G*D
    float* pooled_x      = pooled_causal + (size_t)GG * DD;       // G*D

    // workspace
    float* c2             = (float*)d_ws;                         // K
    float* pooled_counter = c2 + KK;                              // G*D
    int*   indices        = (int*)(pooled_counter + (size_t)GG * DD); // N

    c2_kernel<<<(KK + 127) / 128, 128, 0, stream>>>(causal_cb, c2);

    const size_t ldsBytes = (size_t)KK * LDSTRIDE * sizeof(float); // 264 KB
    argmin_kernel<<<NN / 128, 256, ldsBytes, stream>>>(x, causal_cb, c2, indices);

    pool_kernel<<<GG, DD, 0, stream>>>(x, batch, codebook, causal_cb, counter_cb,
                                       indices, z_nodes, pooled_causal, pooled_x,
                                       pooled_counter);

    fc_kernel<<<(GG * TT + 127) / 128, 128, 0, stream>>>(pooled_causal, pooled_counter,
                                                         pooled_x, fc_w, fc_b,
                                                         causal_pre, counter_pre, y_pre);
}